// MIX_56264071578238
// MI455X (gfx1250) — compile-verified
//
#include <hip/hip_runtime.h>

// Problem constants from the reference
#define Bsz 8192
#define Nsz 4096

// Tiling
#define BT 128          // batch rows per block (8 waves x 16)
#define NT 64           // neurons per block
#define S  68          // padded LDS row stride (floats): 16B-aligned float4s, bank-spread
#define ZS (NT * 16)    // zero-slot offset inside lds_W (8 zero floats)

typedef __attribute__((ext_vector_type(2))) float v2f;
typedef __attribute__((ext_vector_type(8))) float v8f;

__device__ __forceinline__ float fast_rcp(float x) {
    return __builtin_amdgcn_rcpf(x);   // v_rcp_f32, ~1 ulp
}

__global__ __launch_bounds__(256) void mixact_wmma_kernel(
    const float* __restrict__ sg,   // [B, N]
    const float* __restrict__ Wg,   // [N, 4, 4]
    const float* __restrict__ bg,   // [N, 4]
    float* __restrict__ outg)       // [B, N]
{
    __shared__ float lds_s[BT * S];        // s tile, later reused as out tile
    __shared__ float lds_W[NT * 16 + 8];   // W tile + 8-float zero slot
    __shared__ float lds_b[NT * 4];        // bias tile

    const int t  = threadIdx.x;
    const int b0 = blockIdx.y * BT;
    const int n0 = blockIdx.x * NT;

    // ---- stage s tile [BT x NT], coalesced float4 ----
    {
        const int c  = (t & 15) * 4;   // column chunk 0..60
        const int r0 = t >> 4;         // 0..15
        for (int swp = 0; swp < BT / 16; ++swp) {
            const int r = swp * 16 + r0;
            const float4 v = *(const float4*)(sg + (size_t)(b0 + r) * Nsz + n0 + c);
            *(float4*)(lds_s + r * S + c) = v;
        }
    }
    // ---- stage W tile (NT*16 contiguous floats) ----
    {
        const float4 v = *(const float4*)(Wg + (size_t)n0 * 16 + t * 4);
        *(float4*)(lds_W + t * 4) = v;
    }
    if (t < 8) lds_W[ZS + t] = 0.0f;       // zero slot for masked A lanes
    // ---- stage bias tile (NT*4 floats) ----
    if (t < NT) {
        const float4 v = *(const float4*)(bg + (size_t)n0 * 4 + t * 4);
        *(float4*)(lds_b + t * 4) = v;
    }
    __syncthreads();

    const int lane = t & 31;
    const int m    = lane & 15;            // A row / B-D column (batch)
    const int h    = lane >> 4;            // lane half -> K index pair {2h, 2h+1}
    const int wrow = (t >> 5) * 16;        // this wave's 16 exclusive s-tile rows
    const int srow = (wrow + m) * S;

    for (int n = 0; n < NT; n += 2) {
        // ---- B fragments = activations, 4(K) x 16(batch), for neurons n and n+1
        //      VGPR0: K=0 (lanes 0-15) / K=2 (lanes 16-31); VGPR1: K=1 / K=3
        const float sv1 = lds_s[srow + n];
        const float sv2 = lds_s[srow + n + 1];
        const float sg1 = fast_rcp(1.0f + __expf(-sv1));
        const float sg2 = fast_rcp(1.0f + __expf(-sv2));
        v2f Bf1, Bf2;
        Bf1.x = h ? fmaxf(sv1, 0.0f) : fminf(sv1, 0.0f);   // relu : antirelu
        Bf1.y = h ? sg1 : sv1;                              // sigmoid : identity
        Bf2.x = h ? fmaxf(sv2, 0.0f) : fminf(sv2, 0.0f);
        Bf2.y = h ? sg2 : sv2;

        // ---- A fragments = W^T: neuron n in rows 0-3, neuron n+1 in rows 8-11
        //      A[row, k] = W[neuron, k, row];  VGPR0 -> k=2h, VGPR1 -> k=2h+1
        const int base1 = n * 16 + 8 * h;                   // W[n, 2h, *]
        const int base2 = (n + 1) * 16 + 8 * h;             // W[n+1, 2h, *]
        const int i1 = (m < 4)             ? (base1 + m)       : ZS;
        const int i2 = (m >= 8 && m < 12)  ? (base2 + (m - 8)) : ZS;
        v2f Af1, Af2;
        Af1.x = lds_W[i1];
        Af1.y = lds_W[i1 + 4];   // k=2h+1 (zero slot is 8 wide, stays zero)
        Af2.x = lds_W[i2];
        Af2.y = lds_W[i2 + 4];

        // ---- chained WMMA: D[v][lane<16]  = logits[b=lane,    n,   j=v]
        //                    D[v][lane>=16] = logits[b=lane-16, n+1, j=v]
        v8f acc = {0.f, 0.f, 0.f, 0.f, 0.f, 0.f, 0.f, 0.f};
        acc = __builtin_amdgcn_wmma_f32_16x16x4_f32(
            false, Af1, false, Bf1, (short)0, acc, false, false);
        acc = __builtin_amdgcn_wmma_f32_16x16x4_f32(
            false, Af2, false, Bf2, (short)0, acc, false, false);

        // ---- per-lane epilogue: each lane owns one (batch, neuron) element
        const int nn = n + h;
        const float4 bb = *(const float4*)(lds_b + nn * 4);
        const float l0 = acc[0] + bb.x;
        const float l1 = acc[1] + bb.y;
        const float l2 = acc[2] + bb.z;
        const float l3 = acc[3] + bb.w;
        const float e0 = __expf(l0);
        const float e1 = __expf(l1);
        const float e2 = __expf(l2);
        const float e3 = __expf(l3);
        const float es = e0 + e1 + e2 + e3;
        const float sv  = h ? sv2 : sv1;
        const float sig = h ? sg2 : sg1;
        const float num = e0 * fminf(sv, 0.0f) + e1 * sv +
                          e2 * fmaxf(sv, 0.0f) + e3 * sig;
        lds_s[srow + nn] = num * fast_rcp(es);  // overwrite consumed s column
    }

    // ---- coalesced writeback: each wave streams out its 16 rows (256B/row) ----
    {
        const int c4 = m * 4;
        for (int it = 0; it < 8; ++it) {
            const int r = wrow + it * 2 + h;
            const float4 v = *(const float4*)(lds_s + r * S + c4);
            *(float4*)(outg + (size_t)(b0 + r) * Nsz + n0 + c4) = v;
        }
    }
}

extern "C" void kernel_launch(void* const* d_in, const int* in_sizes, int n_in,
                              void* d_out, int out_size, void* d_ws, size_t ws_size,
                              hipStream_t stream) {
    const float* s = (const float*)d_in[0];   // [8192, 4096]
    const float* W = (const float*)d_in[1];   // [4096, 4, 4]
    const float* b = (const float*)d_in[2];   // [4096, 4]
    float* out = (float*)d_out;               // [8192, 4096]

    dim3 grid(Nsz / NT, Bsz / BT);            // (64, 64)
    mixact_wmma_kernel<<<grid, 256, 0, stream>>>(s, W, b, out);
}